// AdditiveAttention_32633161515535
// MI455X (gfx1250) — compile-verified
//
#include <hip/hip_runtime.h>
#include <math.h>

typedef float v2f __attribute__((ext_vector_type(2)));
typedef float v8f __attribute__((ext_vector_type(8)));

#define B_   256
#define S_   2048
#define H_   128
#define W1P  132   // 128 + 4-float pad: row stride ≡ 4 mod 64 banks -> conflict-free b64 LDS loads

__global__ __launch_bounds__(256)
void addattn_kernel(const float* __restrict__ dec_state,      // [B,H]
                    const float* __restrict__ enc,            // [B,S,H]
                    const unsigned char* __restrict__ mask,   // [B,S] bool
                    const float* __restrict__ W1,             // [H,H]
                    const float* __restrict__ W2,             // [H,H]
                    const float* __restrict__ v,              // [H]
                    float* __restrict__ ctx_out,              // [B,H]
                    float* __restrict__ attn_out)             // [B,S]
{
    __shared__ float w1_lds[H_ * W1P];
    __shared__ float scores[S_];       // scores, then normalized attn
    __shared__ float dec_lds[H_];
    __shared__ float v_lds[H_];
    __shared__ float ctx_red[H_];
    __shared__ float rbuf[16];         // [0..7] wave max, [8..15] wave sum

    const int b    = blockIdx.x;
    const int tid  = threadIdx.x;
    const int lane = tid & 31;
    const int wave = tid >> 5;         // 0..7
    const int half = lane >> 4;        // 0 or 1
    const int l15  = lane & 15;

    // ---- stage W1 into padded LDS (row-major [k][h]) ----
    for (int idx = tid; idx < H_ * H_; idx += 256) {
        int r = idx >> 7, c = idx & 127;
        w1_lds[r * W1P + c] = W1[idx];
    }
    if (tid < H_) v_lds[tid] = v[tid];

    // ---- dec[k] = sum_h decoder_state[b,h] * W2[k,h] ----
    if (tid < H_) {
        const float* dsr = dec_state + (size_t)b * H_;
        const float* w2r = W2 + (size_t)tid * H_;
        float acc = 0.f;
        #pragma unroll 8
        for (int h = 0; h < H_; ++h) acc += dsr[h] * w2r[h];
        dec_lds[tid] = acc;
    }
    __syncthreads();

    // ==== projection GEMM (f32 WMMA) + fused tanh/v-dot epilogue ====
    // Each wave processes s-tiles st = wave, wave+8, ... (16 tiles each, no divergence).
    for (int st = wave; st < S_ / 16; st += 8) {
        const int s0 = st * 16;

        // A fragments: 16x4 f32 per chunk. Lane (half,l15): row=l15, K = 4c + 2*half (+1).
        v2f afrag[32];
        const float* ap = enc + ((size_t)b * S_ + (size_t)(s0 + l15)) * H_ + half * 2;
        #pragma unroll
        for (int c = 0; c < 32; ++c)
            afrag[c] = *(const v2f*)(ap + 4 * c);

        float psum[8];
        #pragma unroll
        for (int m = 0; m < 8; ++m) psum[m] = 0.f;

        for (int kt = 0; kt < 8; ++kt) {
            // Two independent WMMA accumulation chains (even/odd chunks) for ILP.
            v8f acc0 = {};
            v8f acc1 = {};
            // B fragment: B[K,n] = W1[kt*16+n, K]; lane holds column n=l15, K = 4c+2*half (+1)
            const float* bp = &w1_lds[(size_t)(kt * 16 + l15) * W1P + half * 2];
            #pragma unroll
            for (int c = 0; c < 32; c += 2) {
                v2f bf0 = *(const v2f*)(bp + 4 * c);
                v2f bf1 = *(const v2f*)(bp + 4 * (c + 1));
                acc0 = __builtin_amdgcn_wmma_f32_16x16x4_f32(
                           false, afrag[c], false, bf0,
                           (short)0, acc0, false, false);
                acc1 = __builtin_amdgcn_wmma_f32_16x16x4_f32(
                           false, afrag[c + 1], false, bf1,
                           (short)0, acc1, false, false);
            }
            // epilogue: lane owns column n; element m is row (m + 8*half)
            const int   n  = kt * 16 + l15;
            const float dv = dec_lds[n];
            const float vn = v_lds[n];
            #pragma unroll
            for (int m = 0; m < 8; ++m)
                psum[m] += tanhf((acc0[m] + acc1[m]) + dv) * vn;
        }

        // reduce each psum[m] across the 16 lanes of each half-wave
        #pragma unroll
        for (int m = 0; m < 8; ++m) {
            float x = psum[m];
            x += __shfl_xor(x, 1, 32);
            x += __shfl_xor(x, 2, 32);
            x += __shfl_xor(x, 4, 32);
            x += __shfl_xor(x, 8, 32);
            psum[m] = x;
        }
        // lanes 0-7 write rows 0-7, lanes 16-23 write rows 8-15
        const int j = lane & 7;
        float myscore = psum[0];
        #pragma unroll
        for (int m = 1; m < 8; ++m) if (j == m) myscore = psum[m];
        if (l15 < 8) {
            const int s = s0 + half * 8 + j;
            const bool mk = mask[(size_t)b * S_ + s] != 0;
            scores[s] = mk ? myscore : -1e9f;
        }
    }
    __syncthreads();

    // ==== softmax over S ====
    float lmax = -INFINITY;
    for (int s = tid; s < S_; s += 256) lmax = fmaxf(lmax, scores[s]);
    #pragma unroll
    for (int off = 16; off > 0; off >>= 1) lmax = fmaxf(lmax, __shfl_xor(lmax, off, 32));
    if (lane == 0) rbuf[wave] = lmax;
    __syncthreads();
    float gmax = rbuf[0];
    #pragma unroll
    for (int w = 1; w < 8; ++w) gmax = fmaxf(gmax, rbuf[w]);

    float lsum = 0.f;
    for (int s = tid; s < S_; s += 256) {
        float e = expf(scores[s] - gmax);
        scores[s] = e;
        lsum += e;
    }
    #pragma unroll
    for (int off = 16; off > 0; off >>= 1) lsum += __shfl_xor(lsum, off, 32);
    if (lane == 0) rbuf[8 + wave] = lsum;
    __syncthreads();
    float gsum = 0.f;
    #pragma unroll
    for (int w = 0; w < 8; ++w) gsum += rbuf[8 + w];
    const float inv = 1.f / gsum;

    for (int s = tid; s < S_; s += 256) {
        float a = scores[s] * inv;
        scores[s] = a;                                  // keep attn in LDS
        attn_out[(size_t)b * S_ + s] = a;
    }
    __syncthreads();

    // ==== context[b,h] = sum_s attn[s] * enc[b,s,h] (second streaming pass) ====
    const int h = tid & 127;
    const int g = tid >> 7;            // 0 or 1, interleaved over s
    float acc = 0.f;
    const float* ep = enc + (size_t)b * S_ * H_ + h;
    for (int s = g; s < S_; s += 2)
        acc += scores[s] * ep[(size_t)s * H_];
    if (g == 1) ctx_red[h] = acc;
    __syncthreads();
    if (g == 0) ctx_out[(size_t)b * H_ + h] = acc + ctx_red[h];
}

extern "C" void kernel_launch(void* const* d_in, const int* in_sizes, int n_in,
                              void* d_out, int out_size, void* d_ws, size_t ws_size,
                              hipStream_t stream) {
    (void)in_sizes; (void)n_in; (void)out_size; (void)d_ws; (void)ws_size;
    const float*         dec_state = (const float*)d_in[0];
    const float*         enc       = (const float*)d_in[1];
    const unsigned char* src_mask  = (const unsigned char*)d_in[2];
    const float*         W1        = (const float*)d_in[3];
    const float*         W2        = (const float*)d_in[4];
    const float*         v         = (const float*)d_in[5];

    float* out  = (float*)d_out;
    float* ctx  = out;               // [B,H] first in return order
    float* attn = out + B_ * H_;     // [B,S] second

    addattn_kernel<<<B_, 256, 0, stream>>>(dec_state, enc, src_mask, W1, W2, v, ctx, attn);
}